// SpatialFusion_9964324126964
// MI455X (gfx1250) — compile-verified
//
#include <hip/hip_runtime.h>
#include <hip/hip_bf16.h>

// Segmented max over the agent dimension:
//   x:  [1, N_total=32, C=256, H=100, W=252] f32
//   rl: [8] int32 (record_len per segment; sums to 32)
//   out:[8, 256, 100, 252] f32  = per-segment elementwise max
//
// Pure streaming reduction: ~825 MB read + ~206 MB write -> ~44 us at
// 23.3 TB/s HBM. No matmul structure (max is not bilinear), so the optimal
// CDNA5 path is wide non-temporal b128 traffic, not WMMA.

typedef float v4f __attribute__((ext_vector_type(4)));

#define SEG_B    8
#define SEG_C    256
#define SEG_H    100
#define SEG_W    252
#define SEG_CHW  (SEG_C * SEG_H * SEG_W)   // 6,451,200 floats per agent
#define SEG_CHW4 (SEG_CHW / 4)             // 1,612,800 float4 per agent
#define BLOCK    256                        // 8 wave32 waves per workgroup

static __device__ __forceinline__ v4f vmax4(v4f a, v4f b) {
    v4f r;
    r.x = fmaxf(a.x, b.x);
    r.y = fmaxf(a.y, b.y);
    r.z = fmaxf(a.z, b.z);
    r.w = fmaxf(a.w, b.w);
    return r;
}

__global__ __launch_bounds__(BLOCK)
void SpatialFusion_segmax_kernel(const v4f* __restrict__ x,
                                 const int* __restrict__ rl,
                                 v4f* __restrict__ out) {
    const int b = blockIdx.y;  // segment id (wave-uniform -> scalar path)

    // Prefix sum of record_len for this segment. rl is 8 entries; the address
    // stream is uniform so these lower to scalar loads + SALU.
    int start = 0;
    int cnt   = 0;
#pragma unroll
    for (int i = 0; i < SEG_B; ++i) {
        const int v = rl[i];
        if (i < b)  start += v;
        if (i == b) cnt = v;
    }

    const long idx = (long)blockIdx.x * BLOCK + threadIdx.x;
    if (idx >= SEG_CHW4) return;

    const v4f* p = x + (long)start * SEG_CHW4 + idx;

    v4f acc = { -INFINITY, -INFINITY, -INFINITY, -INFINITY };

    // Agents unrolled x2: two global_load_b128 (non-temporal) in flight.
    int j = 0;
    for (; j + 2 <= cnt; j += 2) {
        const v4f a0 = __builtin_nontemporal_load(p);
        const v4f a1 = __builtin_nontemporal_load(p + SEG_CHW4);
        p += 2L * SEG_CHW4;
        acc = vmax4(acc, a0);
        acc = vmax4(acc, a1);
    }
    if (j < cnt) {
        const v4f a0 = __builtin_nontemporal_load(p);
        acc = vmax4(acc, a0);
    }

    // Streaming store; output is written exactly once.
    __builtin_nontemporal_store(acc, out + (long)b * SEG_CHW4 + idx);
}

extern "C" void kernel_launch(void* const* d_in, const int* in_sizes, int n_in,
                              void* d_out, int out_size, void* d_ws, size_t ws_size,
                              hipStream_t stream) {
    (void)in_sizes; (void)n_in; (void)out_size; (void)d_ws; (void)ws_size;

    const v4f* x  = (const v4f*)d_in[0];   // f32 [1,32,256,100,252] viewed as float4
    const int* rl = (const int*)d_in[1];   // int32 [8]
    v4f*       o  = (v4f*)d_out;           // f32 [8,256,100,252] viewed as float4

    dim3 grid((SEG_CHW4 + BLOCK - 1) / BLOCK, SEG_B, 1);  // 6300 x 8 blocks
    dim3 block(BLOCK, 1, 1);
    SpatialFusion_segmax_kernel<<<grid, block, 0, stream>>>(x, rl, o);
}